// RelevancePropagationAdaptiveAvgPool2d_22076131901613
// MI455X (gfx1250) — compile-verified
//
#include <hip/hip_runtime.h>
#include <math.h>

// LRP m-rule backward through AdaptiveAvgPool2d(7) on x=(32,256,56,56) f32.
// Memory-bound (~206 MB @ 23.3 TB/s ~ 9us). One wave32 handles 4 "window-rows"
// (4 x 448 contiguous floats = 28 windows of 8x8); each LANE owns one full
// window -> all reductions are lane-local, zero cross-lane ops. Data staged
// global->LDS with CDNA5 async copies, results drained LDS->global async.

#ifndef __has_builtin
#define __has_builtin(x) 0
#endif

// b64 async ops move one int2 per lane; builtins take typed AS(1)/AS(3) pointers
typedef int v2i __attribute__((ext_vector_type(2)));
typedef __attribute__((address_space(1))) v2i* gptr_b64;   // global
typedef __attribute__((address_space(3))) v2i* lptr_b64;   // LDS

#if __has_builtin(__builtin_amdgcn_global_load_async_to_lds_b64) && \
    __has_builtin(__builtin_amdgcn_global_store_async_from_lds_b64)
#define USE_ASYNC_LDS 1
#else
#define USE_ASYNC_LDS 0
#pragma message("gfx1250 async to/from LDS builtins NOT available; falling back to plain loads")
#endif

#if __has_builtin(__builtin_amdgcn_s_wait_asynccnt)
#define WAIT_ASYNC0() __builtin_amdgcn_s_wait_asynccnt(0)
#else
#define WAIT_ASYNC0() asm volatile("s_wait_asynccnt 0" ::: "memory")
#endif

#if __has_builtin(__builtin_amdgcn_s_wait_dscnt)
#define WAIT_DS0() __builtin_amdgcn_s_wait_dscnt(0)
#else
#define WAIT_DS0() asm volatile("s_wait_dscnt 0" ::: "memory")
#endif

#define EPSF      1e-5f
#define K_INV     (1.0f / 64.0f)
// nw element: (1/64) / (sqrt(64*(1/64)^2) + eps) = (1/64)/(0.125 + 1e-5)
#define NW_C      ((1.0f / 64.0f) / (0.125f + 1e-5f))
#define SIN_T     (1.2246467991473532e-16f)

// padded window-row stride in LDS words: 448 + 4. 452/4 = 113 == 1 (mod 16),
// so per-lane b128 window reads spread across bank-quads (<=2-way).
#define WR_STRIDE 452

__global__ __launch_bounds__(256) void lrp_avgpool_bwd_kernel(
    const float* __restrict__ x, const float* __restrict__ r,
    float* __restrict__ out, int n_wrows)
{
    // 8 waves per block, 4 window-rows per wave
    __shared__ float lds[8 * 4 * WR_STRIDE];

    const int lane = threadIdx.x & 31;
    const int wv   = threadIdx.x >> 5;
    const int wr0  = (blockIdx.x * 8 + wv) * 4;     // first window-row (wave-uniform)
    if (wr0 >= n_wrows) return;                     // uniform per wave; no barriers used

    float* wbuf = &lds[wv * (4 * WR_STRIDE)];

    // ---- stage 4 window-rows into LDS: 7 x b64 per lane per subrow, coalesced
#if USE_ASYNC_LDS
    #pragma unroll
    for (int sr = 0; sr < 4; ++sr) {
        const float* g = x + (size_t)(wr0 + sr) * 448 + lane * 2;
        float*       l = wbuf + sr * WR_STRIDE + lane * 2;
        #pragma unroll
        for (int i = 0; i < 7; ++i) {
            __builtin_amdgcn_global_load_async_to_lds_b64(
                (gptr_b64)(g + i * 64), (lptr_b64)(l + i * 64), 0, 0);
        }
    }
#else
    #pragma unroll
    for (int sr = 0; sr < 4; ++sr) {
        const float* g = x + (size_t)(wr0 + sr) * 448 + lane * 2;
        float*       l = wbuf + sr * WR_STRIDE + lane * 2;
        #pragma unroll
        for (int i = 0; i < 7; ++i)
            *(float2*)(l + i * 64) = *(const float2*)(g + i * 64);
    }
#endif

#if USE_ASYNC_LDS
    WAIT_ASYNC0();          // ASYNCcnt is per-wave; whole 4-row tile now in LDS
#endif

    // ---- each lane < 28 owns one full 8x8 window: window id = wr0*7 + lane
    if (lane < 28) {
        const int subrow = (lane * 37) >> 8;        // lane / 7  (lane <= 27)
        const int ox     = lane - subrow * 7;       // lane % 7
        float* wnd = wbuf + subrow * WR_STRIDE + ox * 8;

        const float rv = r[(size_t)wr0 * 7 + lane]; // coalesced across lanes

        // pass 1: load 64 elems (16 x ds_load_b128), accumulate sum / sumsq
        float4 xs[16];
        float sumx = 0.0f, sumsq = 0.0f;
        #pragma unroll
        for (int rr = 0; rr < 8; ++rr) {
            float4 a = *(float4*)(wnd + rr * 56);
            float4 b = *(float4*)(wnd + rr * 56 + 4);
            xs[rr * 2 + 0] = a;
            xs[rr * 2 + 1] = b;
            sumx  += a.x + a.y + a.z + a.w + b.x + b.y + b.z + b.w;
            sumsq = fmaf(a.x, a.x, sumsq); sumsq = fmaf(a.y, a.y, sumsq);
            sumsq = fmaf(a.z, a.z, sumsq); sumsq = fmaf(a.w, a.w, sumsq);
            sumsq = fmaf(b.x, b.x, sumsq); sumsq = fmaf(b.y, b.y, sumsq);
            sumsq = fmaf(b.z, b.z, sumsq); sumsq = fmaf(b.w, b.w, sumsq);
        }

        // scalars: with A = sin*norm/no, B = sin*wx/no the LRP m-rule collapses
        // to out_i = (alpha*x_i + beta) * scale  (cos=-1 => x_head=-(1+B)x + A*c)
        const float norm     = sqrtf(sumsq) + EPSF;
        const float inv_norm = 1.0f / norm;
        const float wx = NW_C * sumx * inv_norm;                    // nx . nw
        const float t  = sumsq * inv_norm * inv_norm;               // sum(nx^2)
        // sum(orth^2) = 64c^2 - 2wx^2 + wx^2*t  (>= 0; clamp rounding noise)
        const float oo = fmaxf(64.0f * (NW_C * NW_C) + wx * wx * (t - 2.0f), 0.0f);
        const float no = sqrtf(oo) + EPSF;
        const float A  = SIN_T * norm / no;
        const float B  = SIN_T * wx / no;
        const float alpha = (2.0f + B) * K_INV;                     // wd = alpha*x + beta
        const float beta  = -(A * NW_C) * K_INV;
        const float S     = alpha * sumx + 64.0f * beta;            // sum(wd)
        const float scale = rv / (S + EPSF);
        const float p = alpha * scale, q = beta * scale;

        // pass 2: out_i = fma(p, x_i, q), written back into the same LDS slots
        #pragma unroll
        for (int rr = 0; rr < 8; ++rr) {
            float4 a = xs[rr * 2 + 0];
            float4 b = xs[rr * 2 + 1];
            float4 oa, ob;
            oa.x = fmaf(p, a.x, q); oa.y = fmaf(p, a.y, q);
            oa.z = fmaf(p, a.z, q); oa.w = fmaf(p, a.w, q);
            ob.x = fmaf(p, b.x, q); ob.y = fmaf(p, b.y, q);
            ob.z = fmaf(p, b.z, q); ob.w = fmaf(p, b.w, q);
            *(float4*)(wnd + rr * 56)     = oa;
            *(float4*)(wnd + rr * 56 + 4) = ob;
        }
    }

    // ---- drain results LDS -> global, coalesced
#if USE_ASYNC_LDS
    WAIT_DS0();             // all lanes' ds_stores visible to the async engine
    #pragma unroll
    for (int sr = 0; sr < 4; ++sr) {
        float* g = out + (size_t)(wr0 + sr) * 448 + lane * 2;
        float* l = wbuf + sr * WR_STRIDE + lane * 2;
        #pragma unroll
        for (int i = 0; i < 7; ++i) {
            __builtin_amdgcn_global_store_async_from_lds_b64(
                (gptr_b64)(g + i * 64), (lptr_b64)(l + i * 64), 0, 0);
        }
    }
    // s_endpgm performs an implicit wait-idle; no trailing wait needed
#else
    #pragma unroll
    for (int sr = 0; sr < 4; ++sr) {
        float* g = out + (size_t)(wr0 + sr) * 448 + lane * 2;
        float* l = wbuf + sr * WR_STRIDE + lane * 2;
        #pragma unroll
        for (int i = 0; i < 7; ++i)
            *(float2*)(g + i * 64) = *(float2*)(l + i * 64);
    }
#endif
}

extern "C" void kernel_launch(void* const* d_in, const int* in_sizes, int n_in,
                              void* d_out, int out_size, void* d_ws, size_t ws_size,
                              hipStream_t stream) {
    const float* x = (const float*)d_in[0];   // (32,256,56,56) f32
    const float* r = (const float*)d_in[1];   // (32,256,7,7)  f32
    float* out = (float*)d_out;               // (32,256,56,56) f32

    // window-row = 8 image rows x 56 cols = 448 contiguous floats (7 windows)
    const int n_wrows = in_sizes[0] / 448;    // 32*256*7 = 57344
    const int blocks  = (n_wrows + 31) / 32;  // 8 waves x 4 window-rows per block

    lrp_avgpool_bwd_kernel<<<blocks, 256, 0, stream>>>(x, r, out, n_wrows);
}